// BoundaryPredictor2_88038239633964
// MI455X (gfx1250) — compile-verified
//
#include <hip/hip_runtime.h>

#define B_  4
#define L_  1500
#define D_  512
#define NH_ 8
#define HD_ 64
#define EPS_  1e-8f
#define PEPS_ 1.1920929e-07f

typedef __attribute__((ext_vector_type(2))) float v2f;
typedef __attribute__((ext_vector_type(4))) float v4f;
typedef __attribute__((ext_vector_type(8))) float v8f;
typedef int i4v __attribute__((vector_size(16)));   // matches async-builtin param

// Use gfx1250 async global->LDS copies when the toolchain exposes them
// (also keeps the host compilation pass clean).
#if __has_builtin(__builtin_amdgcn_global_load_async_to_lds_b128) && \
    __has_builtin(__builtin_amdgcn_s_wait_asynccnt)
#define USE_ASYNC 1
#else
#define USE_ASYNC 0
#endif

// Copy 16 bytes global -> LDS (per lane).
__device__ __forceinline__ void cp16(const float* __restrict__ g, float* l)
{
#if USE_ASYNC
    __builtin_amdgcn_global_load_async_to_lds_b128(
        (__attribute__((address_space(1))) i4v*)g,
        (__attribute__((address_space(3))) i4v*)l, 0, 0);
#else
    *(v4f*)l = *(const v4f*)g;
#endif
}

__device__ __forceinline__ void async_fence_block()
{
#if USE_ASYNC
    __builtin_amdgcn_s_wait_asynccnt(0);
#endif
    __syncthreads();
}

// ---------------------------------------------------------------------------
// GEMM: C[M,N] = act( A[M,K] @ W[N,K]^T + bias ), fp32 WMMA 16x16x4.
// Block tile 64x128 (8 waves, 2Mx4N), wave tile 32x32 (4 accumulators:
// 4 WMMA per 4 fragment loads -> 1.5x FLOP/LDS-byte vs 16x32 tiles).
// K staged in 64-wide slabs, double-buffered LDS (stride 68 floats:
// 16B-aligned rows, bank pattern (4m+k)%64 conflict-free fragment reads),
// filled by ASYNCcnt-tracked global_load_async_to_lds_b128 when available.
// ---------------------------------------------------------------------------
#define TM 64
#define TN 128
#define TK 64
#define LSTR (TK + 4)

__global__ __launch_bounds__(256) void gemm_wmma_f32(
    const float* __restrict__ A, const float* __restrict__ W,
    const float* __restrict__ bias, float* __restrict__ C,
    int M, int N, int K, int act)
{
    __shared__ float sA[2 * TM * LSTR];   //  34,816 B
    __shared__ float sW[2 * TN * LSTR];   //  69,632 B (104 KB total)

    const int tid  = threadIdx.x;
    const int wave = tid >> 5;
    const int lane = tid & 31;
    const int half = lane >> 4;   // K-pair (frags) / M+8 (output) selector
    const int l16  = lane & 15;   // M row (A frag) / N col (B frag, output)

    const int tileM = blockIdx.x * TM;
    const int tileN = blockIdx.y * TN;
    const int waveM = (wave & 1) * 32;    // 0,32
    const int waveN = (wave >> 1) * 32;   // 0,32,64,96

    v8f acc00 = {}, acc01 = {}, acc10 = {}, acc11 = {};

    const int nkb = K / TK;

    // Stage one 64-wide K slab (A: TM rows, W: TN rows) into LDS buffer `sel`.
    auto stage = [&](int kb, int sel) {
        float* dA = &sA[sel * TM * LSTR];
        float* dW = &sW[sel * TN * LSTR];
        #pragma unroll
        for (int i = 0; i < 4; ++i) {               // TM*TK/4 = 1024 vec4
            int idx = tid + i * 256;
            int row = idx >> 4;
            int c4  = idx & 15;
            int gr  = min(tileM + row, M - 1);      // clamp: OOB rows unused
            cp16(&A[(size_t)gr * K + kb + c4 * 4], &dA[row * LSTR + c4 * 4]);
        }
        #pragma unroll
        for (int i = 0; i < 8; ++i) {               // TN*TK/4 = 2048 vec4
            int idx = tid + i * 256;
            int row = idx >> 4;
            int c4  = idx & 15;
            int gr  = min(tileN + row, N - 1);
            cp16(&W[(size_t)gr * K + kb + c4 * 4], &dW[row * LSTR + c4 * 4]);
        }
    };

    stage(0, 0);
    async_fence_block();

    int sel = 0;
    for (int kb = 0; kb < nkb; ++kb) {
        if (kb + 1 < nkb) stage((kb + 1) * TK, sel ^ 1);  // prefetch next slab

        const float* cA = &sA[sel * TM * LSTR];
        const float* cW = &sW[sel * TN * LSTR];
        #pragma unroll
        for (int kk = 0; kk < TK; kk += 4) {
            // A 16x4 frag: lane<16 K=kk,kk+1 ; lane>=16 K=kk+2,kk+3
            v2f a0 = *(const v2f*)&cA[(waveM +      l16) * LSTR + kk + 2 * half];
            v2f a1 = *(const v2f*)&cA[(waveM + 16 + l16) * LSTR + kk + 2 * half];
            v2f b0 = *(const v2f*)&cW[(waveN +      l16) * LSTR + kk + 2 * half];
            v2f b1 = *(const v2f*)&cW[(waveN + 16 + l16) * LSTR + kk + 2 * half];
            acc00 = __builtin_amdgcn_wmma_f32_16x16x4_f32(
                false, a0, false, b0, (short)0, acc00, false, false);
            acc01 = __builtin_amdgcn_wmma_f32_16x16x4_f32(
                false, a0, false, b1, (short)0, acc01, false, false);
            acc10 = __builtin_amdgcn_wmma_f32_16x16x4_f32(
                false, a1, false, b0, (short)0, acc10, false, false);
            acc11 = __builtin_amdgcn_wmma_f32_16x16x4_f32(
                false, a1, false, b1, (short)0, acc11, false, false);
        }
        async_fence_block();   // next slab resident + all reads of `sel` done
        sel ^= 1;
    }

    // Epilogue: C/D layout -> VGPR v holds M = v + 8*half, N = l16
    #pragma unroll
    for (int v = 0; v < 8; ++v) {
        int m0 = tileM + waveM + v + 8 * half;
        int m1 = m0 + 16;
        int n0 = tileN + waveN + l16;
        int n1 = n0 + 16;
        float bz0 = bias ? bias[n0] : 0.0f;
        float bz1 = bias ? bias[n1] : 0.0f;
        float x00 = acc00[v] + bz0, x01 = acc01[v] + bz1;
        float x10 = acc10[v] + bz0, x11 = acc11[v] + bz1;
        if (act == 1) {  // exact GELU
            x00 = 0.5f * x00 * (1.0f + erff(x00 * 0.7071067811865475f));
            x01 = 0.5f * x01 * (1.0f + erff(x01 * 0.7071067811865475f));
            x10 = 0.5f * x10 * (1.0f + erff(x10 * 0.7071067811865475f));
            x11 = 0.5f * x11 * (1.0f + erff(x11 * 0.7071067811865475f));
        }
        if (m0 < M) {
            C[(size_t)m0 * N + n0] = x00;
            C[(size_t)m0 * N + n1] = x01;
        }
        if (m1 < M) {
            C[(size_t)m1 * N + n0] = x10;
            C[(size_t)m1 * N + n1] = x11;
        }
    }
}

// ---------------------------------------------------------------------------
// Row-wise L2 normalize: y = x / max(||x||, EPS)    (rows of 512)
// ---------------------------------------------------------------------------
__global__ __launch_bounds__(256) void rownorm_kernel(
    const float* __restrict__ x, float* __restrict__ y)
{
    __shared__ float red[256];
    int r = blockIdx.x, t = threadIdx.x;
    const float* xr = &x[(size_t)r * D_];
    float a = xr[t], b = xr[t + 256];
    red[t] = a * a + b * b;
    __syncthreads();
    for (int s = 128; s > 0; s >>= 1) {
        if (t < s) red[t] += red[t + s];
        __syncthreads();
    }
    float inv = 1.0f / fmaxf(sqrtf(red[0]), EPS_);
    float* yr = &y[(size_t)r * D_];
    yr[t] = a * inv;
    yr[t + 256] = b * inv;
}

// y = nrm(p + q) row-wise (in-place safe: each block owns its row)
__global__ __launch_bounds__(256) void resnorm_kernel(
    const float* __restrict__ p, const float* __restrict__ q,
    float* __restrict__ y)
{
    __shared__ float red[256];
    int r = blockIdx.x, t = threadIdx.x;
    size_t o = (size_t)r * D_;
    float a = p[o + t] + q[o + t];
    float b = p[o + t + 256] + q[o + t + 256];
    red[t] = a * a + b * b;
    __syncthreads();
    for (int s = 128; s > 0; s >>= 1) {
        if (t < s) red[t] += red[t + s];
        __syncthreads();
    }
    float inv = 1.0f / fmaxf(sqrtf(red[0]), EPS_);
    y[o + t] = a * inv;
    y[o + t + 256] = b * inv;
}

// LayerNorm (rows of 512), eps 1e-5
__global__ __launch_bounds__(256) void layernorm_kernel(
    const float* __restrict__ x, const float* __restrict__ g,
    const float* __restrict__ bl, float* __restrict__ y)
{
    __shared__ float red[256];
    int r = blockIdx.x, t = threadIdx.x;
    const float* xr = &x[(size_t)r * D_];
    float a = xr[t], b = xr[t + 256];
    red[t] = a + b;
    __syncthreads();
    for (int s = 128; s > 0; s >>= 1) {
        if (t < s) red[t] += red[t + s];
        __syncthreads();
    }
    float mu = red[0] * (1.0f / 512.0f);
    __syncthreads();
    float ca = a - mu, cb = b - mu;
    red[t] = ca * ca + cb * cb;
    __syncthreads();
    for (int s = 128; s > 0; s >>= 1) {
        if (t < s) red[t] += red[t + s];
        __syncthreads();
    }
    float rs = rsqrtf(red[0] * (1.0f / 512.0f) + 1e-5f);
    float* yr = &y[(size_t)r * D_];
    yr[t]       = ca * rs * g[t] + bl[t];
    yr[t + 256] = cb * rs * g[t + 256] + bl[t + 256];
}

// ---------------------------------------------------------------------------
// Adjacent-frame cosine -> boundary prob -> RelaxedBernoulli hard sample
// ---------------------------------------------------------------------------
__global__ __launch_bounds__(128) void boundary_kernel(
    const float* __restrict__ qf, const float* __restrict__ kf,
    const float* __restrict__ u_noise, const float* __restrict__ sim_bias,
    float* __restrict__ hard)
{
    int b = blockIdx.x / L_, l = blockIdx.x % L_;
    int t = threadIdx.x;
    __shared__ float red[128];
    float probs = 0.0f;  // padded last column
    if (l < L_ - 1) {
        const float* q = &qf[((size_t)b * L_ + l) * D_];
        const float* k = &kf[((size_t)b * L_ + l + 1) * D_];
        float s = 0.0f;
        for (int i = t; i < D_; i += 128) s += q[i] * k[i];
        red[t] = s;
        __syncthreads();
        for (int ss = 64; ss > 0; ss >>= 1) {
            if (t < ss) red[t] += red[t + ss];
            __syncthreads();
        }
        float c = red[0];
        probs = fminf(fmaxf((1.0f - (c + sim_bias[0])) * 0.5f, 0.0f), 1.0f);
    }
    if (t == 0) {
        float p = fminf(fmaxf(probs, PEPS_), 1.0f - PEPS_);
        float u = fminf(fmaxf(u_noise[b * L_ + l], PEPS_), 1.0f - PEPS_);
        float z = logf(p) - log1pf(-p) + logf(u) - log1pf(-u);  // TEMP = 1
        float soft = 1.0f / (1.0f + expf(-z));
        hard[b * L_ + l] = (soft > 0.5f) ? 1.0f : 0.0f;
    }
}

// ---------------------------------------------------------------------------
// Per-batch serial scan: length masks, emergency boundary, segment starts
// ---------------------------------------------------------------------------
__global__ void scan_kernel(const float* __restrict__ lengths,
                            float* __restrict__ hard,
                            int* __restrict__ segstart,
                            int* __restrict__ nseg)
{
    int b = blockIdx.x;
    if (threadIdx.x != 0) return;
    float len = lengths[b];
    int valid_len = min((int)(len * (float)(L_ + 1)) - 1, L_);
    bool trunc = (valid_len < L_);
    float sum = 0.0f;
    for (int l = 0; l < L_; ++l) {
        float h = hard[b * L_ + l];
        if (trunc && l >= valid_len) h = (l == valid_len) ? 1.0f : 0.0f;
        hard[b * L_ + l] = h;
        sum += h;
    }
    bool no_b = (sum == 0.0f);
    int e = min(valid_len, L_ - 1);
    int cum = 0;
    segstart[b * (L_ + 1)] = 0;
    for (int l = 0; l < L_; ++l) {
        float h = hard[b * L_ + l];
        if (no_b && l == e) h = 1.0f;
        if (h > 0.5f) {
            ++cum;
            if (cum <= L_) segstart[b * (L_ + 1) + cum] = l + 1;
        }
    }
    // number of non-empty segments = segid of last token + 1
    int last_h = (hard[b * L_ + L_ - 1] > 0.5f) ||
                 (no_b && e == L_ - 1) ? 1 : 0;
    int ns = cum - last_h + 1;
    nseg[b] = ns;
    segstart[b * (L_ + 1) + ns] = L_;
}

// ---------------------------------------------------------------------------
// base[b,h,l] = (q_h . keys[b,l,h,:]) / sqrt(64)
// ---------------------------------------------------------------------------
__global__ __launch_bounds__(256) void basescore_kernel(
    const float* __restrict__ keys, const float* __restrict__ lq,
    float* __restrict__ base)
{
    int bl = blockIdx.x;
    int b = bl / L_, l = bl % L_;
    int t = threadIdx.x;
    int h = t >> 5, lane = t & 31;
    const float* kp = &keys[((size_t)b * L_ + l) * D_ + h * HD_];
    const float* qp = &lq[h * HD_];
    float s = qp[lane] * kp[lane] + qp[lane + 32] * kp[lane + 32];
    s += __shfl_xor(s, 16, 32);
    s += __shfl_xor(s, 8, 32);
    s += __shfl_xor(s, 4, 32);
    s += __shfl_xor(s, 2, 32);
    s += __shfl_xor(s, 1, 32);
    if (lane == 0) base[((size_t)b * NH_ + h) * L_ + l] = s * 0.125f;
}

// ---------------------------------------------------------------------------
// Segmented softmax-pool: pooled[b,s,h,:] = softmax(base over segment) @ vals
// Empty / out-of-range segments -> zeros (matches nan_to_num of -inf softmax)
// ---------------------------------------------------------------------------
__global__ __launch_bounds__(512) void segpool_kernel(
    const float* __restrict__ base, const float* __restrict__ vals,
    const int* __restrict__ segstart, const int* __restrict__ nseg,
    const float* __restrict__ lengths, float* __restrict__ pooled)
{
    int bs = blockIdx.x;
    int b = bs / L_, s = bs % L_;
    int t = threadIdx.x;
    int h = t >> 6, d = t & 63;
    float out = 0.0f;
    if (s < nseg[b]) {
        int st = segstart[b * (L_ + 1) + s];
        int en = segstart[b * (L_ + 1) + s + 1];
        int lm = (int)(lengths[b] * (float)L_);
        en = min(en, lm);
        if (st < en) {
            const float* bp = &base[((size_t)b * NH_ + h) * L_];
            float m = -3.4e38f;
            for (int l = st; l < en; ++l) m = fmaxf(m, bp[l]);
            float sum = 0.0f, acc = 0.0f;
            for (int l = st; l < en; ++l) {
                float e = expf(bp[l] - m);
                sum += e;
                acc += e * vals[((size_t)b * L_ + l) * D_ + h * HD_ + d];
            }
            out = acc / sum;
        }
    }
    pooled[((size_t)b * L_ + s) * D_ + h * HD_ + d] = out;
}

// ---------------------------------------------------------------------------
extern "C" void kernel_launch(void* const* d_in, const int* in_sizes, int n_in,
                              void* d_out, int out_size, void* d_ws, size_t ws_size,
                              hipStream_t stream)
{
    const float* hidden  = (const float*)d_in[0];
    const float* lengths = (const float*)d_in[1];
    const float* u_noise = (const float*)d_in[2];
    const float* W1      = (const float*)d_in[3];
    const float* b1      = (const float*)d_in[4];
    const float* W2      = (const float*)d_in[5];
    const float* b2      = (const float*)d_in[6];
    const float* Wq      = (const float*)d_in[7];
    const float* Wk      = (const float*)d_in[8];
    const float* simb    = (const float*)d_in[9];
    const float* lq      = (const float*)d_in[10];
    const float* Wpk     = (const float*)d_in[11];
    const float* Wpv     = (const float*)d_in[12];
    const float* Wpo     = (const float*)d_in[13];
    const float* ln_g    = (const float*)d_in[14];
    const float* ln_b    = (const float*)d_in[15];
    float* out = (float*)d_out;

    const int M = B_ * L_;                       // 6000
    const size_t BIG = (size_t)M * D_ * sizeof(float);

    char* ws = (char*)d_ws;
    float* buf0 = (float*)ws;  ws += BIG;        // nhid -> hn
    float* buf1 = (float*)ws;  ws += BIG;        // gelu -> qf -> vals
    float* buf2 = (float*)ws;  ws += BIG;        // mlp_out/r -> keys
    float* buf3 = (float*)ws;  ws += BIG;        // kf -> pooled
    float* hard = (float*)ws;  ws += (size_t)B_ * L_ * sizeof(float);
    float* base = (float*)ws;  ws += (size_t)B_ * NH_ * L_ * sizeof(float);
    int* segstart = (int*)ws;  ws += (size_t)B_ * (L_ + 1) * sizeof(int);
    int* nseg = (int*)ws;

    dim3 gg((M + TM - 1) / TM, D_ / TN);         // (94, 4)

    // 1) nhid = nrm(hidden) (row-wise, shared by q_in / k_in slices)
    rownorm_kernel<<<M, 256, 0, stream>>>(hidden, buf0);
    // 2) MLP: t1 = gelu(nhid @ W1^T + b1) ; r0 = t1 @ W2^T + b2
    gemm_wmma_f32<<<gg, 256, 0, stream>>>(buf0, W1, b1, buf1, M, D_, D_, 1);
    gemm_wmma_f32<<<gg, 256, 0, stream>>>(buf1, W2, b2, buf2, M, D_, D_, 0);
    // 3) r = nrm(r0 + nhid)
    resnorm_kernel<<<M, 256, 0, stream>>>(buf2, buf0, buf2);
    // 4) qf = r @ Wq^T ; kf = r @ Wk^T
    gemm_wmma_f32<<<gg, 256, 0, stream>>>(buf2, Wq, nullptr, buf1, M, D_, D_, 0);
    gemm_wmma_f32<<<gg, 256, 0, stream>>>(buf2, Wk, nullptr, buf3, M, D_, D_, 0);
    // 5) boundaries + per-batch scan
    boundary_kernel<<<M, 128, 0, stream>>>(buf1, buf3, u_noise, simb, hard);
    scan_kernel<<<B_, 32, 0, stream>>>(lengths, hard, segstart, nseg);
    // 6) hn = layernorm(hidden); keys/vals projections
    layernorm_kernel<<<M, 256, 0, stream>>>(hidden, ln_g, ln_b, buf0);
    gemm_wmma_f32<<<gg, 256, 0, stream>>>(buf0, Wpk, nullptr, buf2, M, D_, D_, 0);
    gemm_wmma_f32<<<gg, 256, 0, stream>>>(buf0, Wpv, nullptr, buf1, M, D_, D_, 0);
    // 7) scores + segmented softmax pooling
    basescore_kernel<<<M, 256, 0, stream>>>(buf2, lq, base);
    segpool_kernel<<<M, 512, 0, stream>>>(base, buf1, segstart, nseg, lengths, buf3);
    // 8) out = pooled @ Wpo^T
    gemm_wmma_f32<<<gg, 256, 0, stream>>>(buf3, Wpo, nullptr, out, M, D_, D_, 0);
}